// HierarchicalLoss_66907000537200
// MI455X (gfx1250) — compile-verified
//
#include <hip/hip_runtime.h>
#include <math.h>

// HierarchicalLoss for MI455X (gfx1250, wave32).
// Bandwidth-bound scalar reduction: 128 MB fp32 in -> one float out (~5.5 us
// of HBM time at 23.3 TB/s). CDNA5 paths: global_load_async_to_lds_b128
// (ASYNCcnt) tile staging, V_WMMA_F32_16X16X4_F32 as the wave reduction engine.

#define WG   256          // threads per block (8 wave32 waves)
#define RPB  4            // batch rows handled per block
#define NN   4096         // nodes per row (matches reference N)
#define USE_ASYNC_LDS 1

typedef __attribute__((ext_vector_type(2))) float v2f;
typedef __attribute__((ext_vector_type(8))) float v8f;

// Sum the 32 lane values of this wave with one V_WMMA_F32_16X16X4_F32.
// A (16x4) holds the lane values, B = ones, C = 0 -> D[m,n] = rowsum(A,m).
// Summing the 8 D VGPRs per lane + one cross-half shfl_xor(16) yields the
// 32-lane total in every lane (correct for any internal A layout since all
// A elements are summed). Executed by every wave with EXEC all-1s.
__device__ __forceinline__ float wave_wmma_sum(float v) {
    v2f a;    a.x = v;    a.y = 0.0f;
    v2f ones; ones.x = 1.0f; ones.y = 1.0f;
    v8f c = {0.f, 0.f, 0.f, 0.f, 0.f, 0.f, 0.f, 0.f};
    c = __builtin_amdgcn_wmma_f32_16x16x4_f32(false, a, false, ones,
                                              (short)0, c, false, false);
    float s = 0.f;
#pragma unroll
    for (int r = 0; r < 8; ++r) s += c[r];
    s += __shfl_xor(s, 16, 32);
    return s;
}

// Full-block (256-thread) sum; result valid on threadIdx.x == 0.
__device__ __forceinline__ float block_sum(float v, float* wsum) {
    const float s = wave_wmma_sum(v);
    const int lane = threadIdx.x & 31;
    const int wave = threadIdx.x >> 5;
    if (lane == 0) wsum[wave] = s;
    __syncthreads();
    float total = 0.f;
    if (threadIdx.x == 0) {
#pragma unroll
        for (int w = 0; w < WG / 32; ++w) total += wsum[w];
    }
    return total;
}

// One exp per element: e = exp(-|x|) feeds both the stable softplus and the
// sigmoid.  softplus(x) = max(x,0) + log1p(e);  sigmoid = (x>=0 ? 1 : e)/(1+e).
__device__ __forceinline__ float bce_and_sigmoid(float x, float t, float w,
                                                 float& p) {
    const float e  = expf(-fabsf(x));
    const float sp = fmaxf(x, 0.f) + log1pf(e);
    const float r  = 1.0f / (1.0f + e);
    p = (x >= 0.f) ? r : e * r;
    return w * (sp - x * t);
}

__global__ __launch_bounds__(WG) void hloss_main_kernel(
        const float* __restrict__ outputs,   // [B, NN]
        const float* __restrict__ targets,   // [B, NN]
        const float* __restrict__ level_w,   // [NN]
        const int*   __restrict__ esrc,      // [E] parent
        const int*   __restrict__ edst,      // [E] child
        float*       __restrict__ partials,  // [gridDim.x]
        int E)
{
    __shared__ __align__(16) float shm[RPB * NN];  // x tile -> sigmoid(x) in place
    __shared__ float wsum[WG / 32];

    const int tid = threadIdx.x;
    const long long base = (long long)blockIdx.x * (RPB * NN);
    const float* gsrc = outputs + base;            // RPB rows are contiguous

#if USE_ASYNC_LDS
    {
        // Stage the 64 KB x-tile HBM -> LDS with async b128 copies (ASYNCcnt).
        const unsigned lbase = (unsigned)(size_t)(void*)&shm[0];
        for (int j = tid; j < (RPB * NN) / 4; j += WG) {
            const unsigned laddr = lbase + (unsigned)j * 16u;
            const unsigned long long gaddr =
                (unsigned long long)(size_t)(gsrc + j * 4);
            asm volatile("global_load_async_to_lds_b128 %0, %1, off"
                         :: "v"(laddr), "v"(gaddr) : "memory");
        }
        asm volatile("s_wait_asynccnt 0" ::: "memory");
    }
#else
    for (int j = tid; j < (RPB * NN) / 4; j += WG)
        ((float4*)shm)[j] = ((const float4*)gsrc)[j];
#endif
    __syncthreads();

    // Phase 1: weighted BCE-with-logits, float4 granularity (b128 traffic);
    // sigmoid(x) written back to LDS in place.
    float accb = 0.f;
    {
        const float4* tg4 = (const float4*)(targets + base);
        float4* sm4 = (float4*)shm;
        for (int q = tid; q < (RPB * NN) / 4; q += WG) {
            const float4 x4 = sm4[q];
            const float4 t4 = tg4[q];
            const int n0 = (q * 4) & (NN - 1);          // contiguous within a row
            const float4 w4 = *(const float4*)(level_w + n0);
            float4 p4;
            accb += bce_and_sigmoid(x4.x, t4.x, w4.x, p4.x);
            accb += bce_and_sigmoid(x4.y, t4.y, w4.y, p4.y);
            accb += bce_and_sigmoid(x4.z, t4.z, w4.z, p4.z);
            accb += bce_and_sigmoid(x4.w, t4.w, w4.w, p4.w);
            sm4[q] = p4;
        }
    }
    __syncthreads();

    // Phase 2: hierarchy consistency relu(p[child] - p[parent]); int4 index
    // loads (L2-resident) amortized across RPB rows of LDS-resident probs.
    float accc = 0.f;
    {
        const int E4 = E >> 2;
        const int4* s4p = (const int4*)esrc;
        const int4* d4p = (const int4*)edst;
        for (int q = tid; q < E4; q += WG) {
            const int4 s4 = s4p[q];
            const int4 d4 = d4p[q];
#pragma unroll
            for (int r = 0; r < RPB; ++r) {
                const float* pr = shm + r * NN;
                accc += fmaxf(pr[d4.x] - pr[s4.x], 0.f);
                accc += fmaxf(pr[d4.y] - pr[s4.y], 0.f);
                accc += fmaxf(pr[d4.z] - pr[s4.z], 0.f);
                accc += fmaxf(pr[d4.w] - pr[s4.w], 0.f);
            }
        }
        for (int e = (E4 << 2) + tid; e < E; e += WG) {   // tail (E % 4 != 0)
            const int s = esrc[e];
            const int d = edst[e];
#pragma unroll
            for (int r = 0; r < RPB; ++r)
                accc += fmaxf(shm[r * NN + d] - shm[r * NN + s], 0.f);
        }
    }

    // ALPHA = 1.0, BETA = 0.5 folded here; division by B*N in the final pass.
    const float total = block_sum(accb + 0.5f * accc, wsum);
    if (tid == 0) partials[blockIdx.x] = total;
}

__global__ __launch_bounds__(WG) void hloss_final_kernel(
        const float* __restrict__ partials, int n, float scale,
        float* __restrict__ out)
{
    __shared__ float wsum[WG / 32];
    float v = 0.f;
    for (int i = threadIdx.x; i < n; i += WG) v += partials[i];
    const float total = block_sum(v, wsum);
    if (threadIdx.x == 0) out[0] = total * scale;
}

extern "C" void kernel_launch(void* const* d_in, const int* in_sizes, int n_in,
                              void* d_out, int out_size, void* d_ws, size_t ws_size,
                              hipStream_t stream) {
    (void)n_in; (void)out_size; (void)ws_size;
    const float* outputs = (const float*)d_in[0];
    const float* targets = (const float*)d_in[1];
    const float* level_w = (const float*)d_in[2];
    const int*   esrc    = (const int*)d_in[3];
    const int*   edst    = (const int*)d_in[4];

    const int BN = in_sizes[0];          // B * N
    const int E  = in_sizes[3];
    const int B  = BN / NN;
    const int nblocks = B / RPB;         // 1024 for the reference shapes

    float* partials = (float*)d_ws;      // 4 KB scratch

    hloss_main_kernel<<<nblocks, WG, 0, stream>>>(
        outputs, targets, level_w, esrc, edst, partials, E);

    const float scale = 1.0f / (float)BN;
    hloss_final_kernel<<<1, WG, 0, stream>>>(partials, nblocks, scale,
                                             (float*)d_out);
}